// PLIFCell_45853070852359
// MI455X (gfx1250) — compile-verified
//
#include <hip/hip_runtime.h>
#include <cmath>

// PLIF cell: s_t, v_t = f(x_t, v_prev, log_tau, threshold)
// B = H = 4096, fp32. Pure streaming elementwise -> HBM-bound (~11.5 us floor
// at 23.3 TB/s for 256 MiB of traffic). No matmul => WMMA not applicable;
// optimize the data-movement path instead (b128 NT streams + gfx1250 prefetch).

typedef __attribute__((ext_vector_type(4))) float floatx4;

#define B_DIM 4096
#define H_DIM 4096
#define HG    (H_DIM / 4)      // 1024 float4 column-groups per row
#define ROWS_PER_THREAD 4
#define BLOCK 256              // 8 wave32 waves per workgroup

// ---------------------------------------------------------------------------
// Pre-kernel: decay[h] = exp(-1/tau) = exp(-exp(-log_tau[h])), H values once.
// Removes 2 transcendentals per element (33.6M v_exp) from the hot loop.
// ---------------------------------------------------------------------------
__global__ void plif_decay_kernel(const float* __restrict__ log_tau,
                                  float* __restrict__ decay, int H) {
    int h = blockIdx.x * blockDim.x + threadIdx.x;
    if (h < H) {
        decay[h] = expf(-expf(-log_tau[h]));
    }
}

// ---------------------------------------------------------------------------
// Main kernel: each thread owns one float4 column-group (h4) and walks
// ROWS_PER_THREAD consecutive rows, so decay/threshold are loaded once and
// reused. x/v/s/v_out are streamed with non-temporal b128 ops.
// ---------------------------------------------------------------------------
__global__ __launch_bounds__(BLOCK) void plif_main_kernel(
    const floatx4* __restrict__ x,
    const floatx4* __restrict__ v,
    const floatx4* __restrict__ decay4,
    const floatx4* __restrict__ thr4,
    floatx4* __restrict__ s_out,
    floatx4* __restrict__ v_out)
{
    const unsigned tid   = blockIdx.x * BLOCK + threadIdx.x;
    const unsigned h4    = tid & (HG - 1u);                 // column group
    const unsigned rbase = (tid >> 10) * ROWS_PER_THREAD;   // first row

    // Per-channel params: 32 KB arrays, reused B times -> WGP$/L2 resident.
    const floatx4 d4   = decay4[h4];
    const floatx4 t4   = thr4[h4];
    const floatx4 omd4 = 1.0f - d4;

    // gfx1250 prefetch (global_prefetch_b8): pull the later rows toward the
    // WGP while the earlier rows' loads/FMAs are in flight. No VGPR cost,
    // does not touch LOADcnt.
    #pragma unroll
    for (int r = ROWS_PER_THREAD / 2; r < ROWS_PER_THREAD; ++r) {
        const unsigned idx = (rbase + r) * HG + h4;
        __builtin_prefetch(&x[idx], 0, 0);
        __builtin_prefetch(&v[idx], 0, 0);
    }

    #pragma unroll
    for (int r = 0; r < ROWS_PER_THREAD; ++r) {
        const unsigned idx = (rbase + r) * HG + h4;

        // Streamed once -> non-temporal: don't pollute the 192 MB L2.
        const floatx4 xv = __builtin_nontemporal_load(&x[idx]);
        const floatx4 vv = __builtin_nontemporal_load(&v[idx]);

        floatx4 st, vn;
        #pragma unroll
        for (int c = 0; c < 4; ++c) {
            // v_t = decay*v_prev + (1-decay)*x_t   (v_fma)
            const float vt = fmaf(d4[c], vv[c], omd4[c] * xv[c]);
            // Heaviside forward (surrogate grad only matters in bwd, not here)
            const float sp = (vt >= t4[c]) ? 1.0f : 0.0f;   // v_cmp + cndmask
            st[c] = sp;
            vn[c] = vt - sp * t4[c];                        // reset='subtract'
        }

        __builtin_nontemporal_store(st, &s_out[idx]);
        __builtin_nontemporal_store(vn, &v_out[idx]);
    }
}

// ---------------------------------------------------------------------------
// Launch: inputs in setup_inputs() order: x_t, v_prev, log_tau, threshold.
// d_out = [s_t (16M floats) | v_t (16M floats)]. d_ws holds decay (16 KB).
// ---------------------------------------------------------------------------
extern "C" void kernel_launch(void* const* d_in, const int* in_sizes, int n_in,
                              void* d_out, int out_size, void* d_ws, size_t ws_size,
                              hipStream_t stream) {
    const float* x_t       = (const float*)d_in[0];
    const float* v_prev    = (const float*)d_in[1];
    const float* log_tau   = (const float*)d_in[2];
    const float* threshold = (const float*)d_in[3];

    float* s_out = (float*)d_out;
    float* v_out = s_out + (size_t)B_DIM * H_DIM;
    float* decay = (float*)d_ws;   // 4096 floats, written before read

    plif_decay_kernel<<<(H_DIM + BLOCK - 1) / BLOCK, BLOCK, 0, stream>>>(
        log_tau, decay, H_DIM);

    const unsigned total_threads = (B_DIM / ROWS_PER_THREAD) * HG; // 1,048,576
    plif_main_kernel<<<total_threads / BLOCK, BLOCK, 0, stream>>>(
        (const floatx4*)x_t, (const floatx4*)v_prev,
        (const floatx4*)decay, (const floatx4*)threshold,
        (floatx4*)s_out, (floatx4*)v_out);
}